// GAT_36618891166220
// MI455X (gfx1250) — compile-verified
//
#include <hip/hip_runtime.h>
#include <hip/hip_bf16.h>
#include <math.h>

typedef __attribute__((ext_vector_type(2))) float v2f;
typedef __attribute__((ext_vector_type(8))) float v8f;

#define EPS_BN 1e-5f
#define NEG_SLOPE 0.2f

// ---------- ordered-uint encoding for float atomic max ----------
__device__ __forceinline__ unsigned f2ord(float f) {
    unsigned b = __float_as_uint(f);
    return (b & 0x80000000u) ? ~b : (b | 0x80000000u);
}
__device__ __forceinline__ float ord2f(unsigned u) {
    return (u & 0x80000000u) ? __uint_as_float(u & 0x7FFFFFFFu)
                             : __uint_as_float(~u);
}
#define ORD_NEG_INF 0x007FFFFFu  // f2ord(-inf)

// ---------- fills ----------
__global__ void fill_f32(float* p, float v, size_t n) {
    size_t i = (size_t)blockIdx.x * blockDim.x + threadIdx.x;
    if (i < n) p[i] = v;
}
__global__ void fill_u32(unsigned* p, unsigned v, size_t n) {
    size_t i = (size_t)blockIdx.x * blockDim.x + threadIdx.x;
    if (i < n) p[i] = v;
}

// ---------- WMMA f32 GEMM: C[M,Nc] = A[M,K] * B[K,Nc] ----------
// One wave computes a 16x64 strip: 4 accumulators share one A fragment per
// K-step. Register double-buffering: loads for step k+4 are issued before
// the 4 v_wmma of step k, so wmma waits only on loads a full iteration old.
// Layouts per CDNA5 ISA 7.12.2:
//  A 16x4: lanes 0-15 row M=l: v0=K0, v1=K1 ; lanes 16-31: v0=K2, v1=K3
//  B 4x16: lanes 0-15 col N=l: v0=K0, v1=K1 ; lanes 16-31: v0=K2, v1=K3
//  C/D 16x16: VGPR r -> row m0 + r + 8*(lane>=16), col n0 + (lane&15)
__global__ __launch_bounds__(256) void wmma_gemm_f32(
    const float* __restrict__ A, const float* __restrict__ B,
    float* __restrict__ C, int M, int K, int Nc) {
    int lane = threadIdx.x & 31;
    int gw = (int)((blockIdx.x * blockDim.x + threadIdx.x) >> 5);
    int nstrips = Nc >> 6;           // 64-col strips
    int mtiles = M >> 4;
    if (gw >= mtiles * nstrips) return;  // wave-uniform
    int m0 = (gw / nstrips) << 4;
    int n0 = (gw % nstrips) << 6;
    int half = lane >> 4;
    int l = lane & 15;
    const float* __restrict__ arow = A + (size_t)(m0 + l) * K + 2 * half;
    const float* __restrict__ bbase = B + (size_t)(2 * half) * Nc + n0 + l;
    v8f ac0 = {0.f,0.f,0.f,0.f,0.f,0.f,0.f,0.f};
    v8f ac1 = ac0, ac2 = ac0, ac3 = ac0;

    // prologue: load K-step 0
    v2f a_c, b_c0, b_c1, b_c2, b_c3;
    a_c.x = arow[0];
    a_c.y = arow[1];
    b_c0.x = bbase[0];       b_c0.y = bbase[Nc + 0];
    b_c1.x = bbase[16];      b_c1.y = bbase[Nc + 16];
    b_c2.x = bbase[32];      b_c2.y = bbase[Nc + 32];
    b_c3.x = bbase[48];      b_c3.y = bbase[Nc + 48];

    for (int k = 0; k < K - 4; k += 4) {
        // issue next K-step's loads first (overlap with wmma below)
        const float* an = arow + (k + 4);
        const float* bn = bbase + (size_t)(k + 4) * Nc;
        v2f a_n, b_n0, b_n1, b_n2, b_n3;
        a_n.x = an[0];
        a_n.y = an[1];
        b_n0.x = bn[0];       b_n0.y = bn[Nc + 0];
        b_n1.x = bn[16];      b_n1.y = bn[Nc + 16];
        b_n2.x = bn[32];      b_n2.y = bn[Nc + 32];
        b_n3.x = bn[48];      b_n3.y = bn[Nc + 48];

        ac0 = __builtin_amdgcn_wmma_f32_16x16x4_f32(false, a_c, false, b_c0, (short)0, ac0, false, false);
        ac1 = __builtin_amdgcn_wmma_f32_16x16x4_f32(false, a_c, false, b_c1, (short)0, ac1, false, false);
        ac2 = __builtin_amdgcn_wmma_f32_16x16x4_f32(false, a_c, false, b_c2, (short)0, ac2, false, false);
        ac3 = __builtin_amdgcn_wmma_f32_16x16x4_f32(false, a_c, false, b_c3, (short)0, ac3, false, false);

        a_c = a_n;
        b_c0 = b_n0; b_c1 = b_n1; b_c2 = b_n2; b_c3 = b_n3;
    }
    // epilogue: last K-step
    ac0 = __builtin_amdgcn_wmma_f32_16x16x4_f32(false, a_c, false, b_c0, (short)0, ac0, false, false);
    ac1 = __builtin_amdgcn_wmma_f32_16x16x4_f32(false, a_c, false, b_c1, (short)0, ac1, false, false);
    ac2 = __builtin_amdgcn_wmma_f32_16x16x4_f32(false, a_c, false, b_c2, (short)0, ac2, false, false);
    ac3 = __builtin_amdgcn_wmma_f32_16x16x4_f32(false, a_c, false, b_c3, (short)0, ac3, false, false);

    float* crow = C + (size_t)(m0 + half * 8) * Nc + n0 + l;
#pragma unroll
    for (int r = 0; r < 8; ++r) {
        size_t o = (size_t)r * Nc;
        crow[o + 0]  = ac0[r];
        crow[o + 16] = ac1[r];
        crow[o + 32] = ac2[r];
        crow[o + 48] = ac3[r];
    }
}

// ---------- a_s/a_d = <h[n,h,:], att_{src,dst}[h,:]> ; C==64 ----------
__global__ __launch_bounds__(256) void attn_dots(
    const float* __restrict__ h, const float* __restrict__ att_s,
    const float* __restrict__ att_d, float* __restrict__ a_s,
    float* __restrict__ a_d, int N, int H) {
    int wid = (int)((blockIdx.x * blockDim.x + threadIdx.x) >> 5);
    int lane = threadIdx.x & 31;
    if (wid >= N * H) return;
    int head = wid % H;
    const float2* row = (const float2*)(h + (size_t)wid * 64);
    const float2* av = (const float2*)(att_s + head * 64);
    const float2* bv = (const float2*)(att_d + head * 64);
    float2 r = row[lane], a = av[lane], b = bv[lane];
    float s = r.x * a.x + r.y * a.y;
    float d = r.x * b.x + r.y * b.y;
#pragma unroll
    for (int off = 16; off; off >>= 1) {
        s += __shfl_xor(s, off, 32);
        d += __shfl_xor(d, off, 32);
    }
    if (lane == 0) { a_s[wid] = s; a_d[wid] = d; }
}

// ---------- edge pass 1: leaky-relu logits + segment max (u32 ordered) ----------
__global__ __launch_bounds__(256) void edge_logits_max(
    const int* __restrict__ ei, int E, int N, int H,
    const float* __restrict__ a_s, const float* __restrict__ a_d,
    float* __restrict__ elog, unsigned* __restrict__ mord) {
    int e = blockIdx.x * blockDim.x + threadIdx.x;
    int Et = E + N;
    if (e >= Et) return;
    int src = (e < E) ? ei[e] : (e - E);
    int dst = (e < E) ? ei[E + e] : (e - E);
    __builtin_prefetch(&a_s[src * H], 0, 1);
    __builtin_prefetch(&a_d[dst * H], 0, 1);
    for (int hh = 0; hh < H; ++hh) {
        float lg = a_s[src * H + hh] + a_d[dst * H + hh];
        lg = (lg > 0.f) ? lg : NEG_SLOPE * lg;
        elog[(size_t)e * H + hh] = lg;
        atomicMax(&mord[dst * H + hh], f2ord(lg));
    }
}

// ---------- edge pass 2: e = exp(l - m), segment sum ----------
__global__ __launch_bounds__(256) void edge_exp_sum(
    const int* __restrict__ ei, int E, int N, int H,
    const unsigned* __restrict__ mord, float* __restrict__ elog,
    float* __restrict__ ssum) {
    int e = blockIdx.x * blockDim.x + threadIdx.x;
    int Et = E + N;
    if (e >= Et) return;
    int dst = (e < E) ? ei[E + e] : (e - E);
    for (int hh = 0; hh < H; ++hh) {
        float m = ord2f(mord[dst * H + hh]);
        float ev = expf(elog[(size_t)e * H + hh] - m);
        elog[(size_t)e * H + hh] = ev;
        atomicAdd(&ssum[dst * H + hh], ev);
    }
}

// ---------- edge pass 3: out[dst] += h[src] * alpha ; wave per edge ----------
// float4 gathers of h[src] row; scalar f32 atomic scatter into out[dst].
__global__ __launch_bounds__(256) void edge_aggregate(
    const int* __restrict__ ei, int E, int N, int H, int C,
    const float* __restrict__ elog, const float* __restrict__ ssum,
    const float* __restrict__ hlin, float* __restrict__ outb) {
    int wid = (int)((blockIdx.x * blockDim.x + threadIdx.x) >> 5);
    int lane = threadIdx.x & 31;
    int Et = E + N;
    if (wid >= Et) return;
    int e = wid;
    int src = (e < E) ? ei[e] : (e - E);
    int dst = (e < E) ? ei[E + e] : (e - E);
    int HC = H * C;
    const float4* hrow = (const float4*)(hlin + (size_t)src * HC);
    float* orow = outb + (size_t)dst * HC;
    __builtin_prefetch(hrow, 0, 1);
    int nchunk = HC >> 2;  // float4 chunks; 4 | C so chunk stays in one head
    for (int v = lane; v < nchunk; v += 32) {
        int hh = (v << 2) / C;
        float alpha = elog[(size_t)e * H + hh] /
                      (ssum[dst * H + hh] + 1e-16f);
        float4 hv = hrow[v];
        int c0 = v << 2;
        atomicAdd(&orow[c0 + 0], hv.x * alpha);
        atomicAdd(&orow[c0 + 1], hv.y * alpha);
        atomicAdd(&orow[c0 + 2], hv.z * alpha);
        atomicAdd(&orow[c0 + 3], hv.w * alpha);
    }
}

// ---------- epilogue layers 0/1: bias + ELU + BatchNorm(eval), float4 ----------
__global__ void post_elu_bn(const float4* __restrict__ outb,
                            const float4* __restrict__ bias,
                            const float4* __restrict__ g,
                            const float4* __restrict__ be,
                            const float4* __restrict__ mean,
                            const float4* __restrict__ var,
                            float4* __restrict__ dstp, int N, int F) {
    int i = blockIdx.x * blockDim.x + threadIdx.x;
    int F4 = F >> 2;
    if (i >= N * F4) return;
    int j = i % F4;
    float4 v = outb[i], bi = bias[j], gg = g[j], bb = be[j], mm = mean[j], vv = var[j];
    float4 r;
    float t;
    t = v.x + bi.x; t = (t > 0.f) ? t : expm1f(t);
    r.x = gg.x * (t - mm.x) * rsqrtf(vv.x + EPS_BN) + bb.x;
    t = v.y + bi.y; t = (t > 0.f) ? t : expm1f(t);
    r.y = gg.y * (t - mm.y) * rsqrtf(vv.y + EPS_BN) + bb.y;
    t = v.z + bi.z; t = (t > 0.f) ? t : expm1f(t);
    r.z = gg.z * (t - mm.z) * rsqrtf(vv.z + EPS_BN) + bb.z;
    t = v.w + bi.w; t = (t > 0.f) ? t : expm1f(t);
    r.w = gg.w * (t - mm.w) * rsqrtf(vv.w + EPS_BN) + bb.w;
    dstp[i] = r;
}

// ---------- epilogue layer 2: bias + log_softmax over 64 cols ----------
__global__ __launch_bounds__(256) void post_logsoftmax(
    const float* __restrict__ outb, const float* __restrict__ b2,
    float* __restrict__ out, int N) {
    int wid = (int)((blockIdx.x * blockDim.x + threadIdx.x) >> 5);
    int lane = threadIdx.x & 31;
    if (wid >= N) return;
    const float* row = outb + (size_t)wid * 64;
    float v0 = row[lane] + b2[lane];
    float v1 = row[lane + 32] + b2[lane + 32];
    float mx = fmaxf(v0, v1);
#pragma unroll
    for (int off = 16; off; off >>= 1) mx = fmaxf(mx, __shfl_xor(mx, off, 32));
    float s = expf(v0 - mx) + expf(v1 - mx);
#pragma unroll
    for (int off = 16; off; off >>= 1) s += __shfl_xor(s, off, 32);
    float lse = mx + logf(s);
    out[(size_t)wid * 64 + lane] = v0 - lse;
    out[(size_t)wid * 64 + lane + 32] = v1 - lse;
}

// ---------- host orchestration ----------
static void run_edge_stage(const int* ei, int E, int N, int H, int C,
                           const float* a_s, const float* a_d, unsigned* mord,
                           float* ssum, float* elog, const float* hlin,
                           float* outb, hipStream_t stream) {
    int Et = E + N;
    size_t nh = (size_t)N * H;
    size_t nhc = (size_t)N * H * C;
    fill_u32<<<(unsigned)((nh + 255) / 256), 256, 0, stream>>>(mord, ORD_NEG_INF, nh);
    fill_f32<<<(unsigned)((nh + 255) / 256), 256, 0, stream>>>(ssum, 0.f, nh);
    fill_f32<<<(unsigned)((nhc + 255) / 256), 256, 0, stream>>>(outb, 0.f, nhc);
    edge_logits_max<<<(Et + 255) / 256, 256, 0, stream>>>(ei, E, N, H, a_s, a_d, elog, mord);
    edge_exp_sum<<<(Et + 255) / 256, 256, 0, stream>>>(ei, E, N, H, mord, elog, ssum);
    edge_aggregate<<<(Et + 7) / 8, 256, 0, stream>>>(ei, E, N, H, C, elog, ssum, hlin, outb);
}

static void run_gemm(const float* A, const float* B, float* C, int M, int K,
                     int Nc, hipStream_t stream) {
    int waves = (M >> 4) * (Nc >> 6);
    int blocks = (waves + 7) / 8;  // 8 waves (256 threads) per block
    wmma_gemm_f32<<<blocks, 256, 0, stream>>>(A, B, C, M, K, Nc);
}

extern "C" void kernel_launch(void* const* d_in, const int* in_sizes, int n_in,
                              void* d_out, int out_size, void* d_ws, size_t ws_size,
                              hipStream_t stream) {
    const float* x        = (const float*)d_in[0];
    const int*   ei       = (const int*)d_in[1];
    const float* W0       = (const float*)d_in[2];
    const float* att_s0   = (const float*)d_in[3];
    const float* att_d0   = (const float*)d_in[4];
    const float* b0       = (const float*)d_in[5];
    const float* g0       = (const float*)d_in[6];
    const float* be0      = (const float*)d_in[7];
    const float* m0       = (const float*)d_in[8];
    const float* v0       = (const float*)d_in[9];
    const float* W1       = (const float*)d_in[10];
    const float* att_s1   = (const float*)d_in[11];
    const float* att_d1   = (const float*)d_in[12];
    const float* b1       = (const float*)d_in[13];
    const float* g1       = (const float*)d_in[14];
    const float* be1      = (const float*)d_in[15];
    const float* m1       = (const float*)d_in[16];
    const float* v1       = (const float*)d_in[17];
    const float* W2       = (const float*)d_in[18];
    const float* att_s2   = (const float*)d_in[19];
    const float* att_d2   = (const float*)d_in[20];
    const float* b2       = (const float*)d_in[21];

    const int IN_C = 128, H = 4, C = 64, HC = 256, OUT_C = 64;
    const int N = in_sizes[0] / IN_C;       // 10000
    const int E = in_sizes[1] / 2;          // 320000

    // workspace layout (all f32 / u32)
    float* buf0 = (float*)d_ws;                    // N*256 : layer input
    float* buf1 = buf0 + (size_t)N * HC;           // N*256 : GEMM output h
    float* buf2 = buf1 + (size_t)N * HC;           // N*256 : aggregation output
    float* a_s  = buf2 + (size_t)N * HC;           // N*4
    float* a_d  = a_s + (size_t)N * H;             // N*4
    unsigned* mord = (unsigned*)(a_d + (size_t)N * H);  // N*4
    float* ssum = (float*)(mord + (size_t)N * H);  // N*4
    float* elog = ssum + (size_t)N * H;            // (E+N)*4

    // ================= layer 0: 128 -> 4x64 concat =================
    {
        run_gemm(x, W0, buf1, N, IN_C, HC, stream);
        int waves = N * H;
        attn_dots<<<(waves + 7) / 8, 256, 0, stream>>>(buf1, att_s0, att_d0, a_s, a_d, N, H);
        run_edge_stage(ei, E, N, H, C, a_s, a_d, mord, ssum, elog, buf1, buf2, stream);
        int n4 = N * (HC >> 2);
        post_elu_bn<<<(n4 + 255) / 256, 256, 0, stream>>>(
            (const float4*)buf2, (const float4*)b0, (const float4*)g0,
            (const float4*)be0, (const float4*)m0, (const float4*)v0,
            (float4*)buf0, N, HC);
    }

    // ================= layer 1: 256 -> 4x64 concat =================
    {
        run_gemm(buf0, W1, buf1, N, HC, HC, stream);
        int waves = N * H;
        attn_dots<<<(waves + 7) / 8, 256, 0, stream>>>(buf1, att_s1, att_d1, a_s, a_d, N, H);
        run_edge_stage(ei, E, N, H, C, a_s, a_d, mord, ssum, elog, buf1, buf2, stream);
        int n4 = N * (HC >> 2);
        post_elu_bn<<<(n4 + 255) / 256, 256, 0, stream>>>(
            (const float4*)buf2, (const float4*)b1, (const float4*)g1,
            (const float4*)be1, (const float4*)m1, (const float4*)v1,
            (float4*)buf0, N, HC);
    }

    // ================= layer 2: 256 -> 64, heads=1 =================
    {
        run_gemm(buf0, W2, buf1, N, HC, OUT_C, stream);
        attn_dots<<<(N + 7) / 8, 256, 0, stream>>>(buf1, att_s2, att_d2, a_s, a_d, N, 1);
        run_edge_stage(ei, E, N, 1, OUT_C, a_s, a_d, mord, ssum, elog, buf1, buf2, stream);
        post_logsoftmax<<<(N + 7) / 8, 256, 0, stream>>>(buf2, b2, (float*)d_out, N);
    }
}